// ShiftedConv2d_39651138076852
// MI455X (gfx1250) — compile-verified
//
#include <hip/hip_runtime.h>
#include <stdint.h>

typedef __attribute__((ext_vector_type(2))) float v2f;
typedef __attribute__((ext_vector_type(8))) float v8f;
typedef __attribute__((ext_vector_type(4))) unsigned int u32x4;
typedef __attribute__((ext_vector_type(4))) int i32x4;
typedef __attribute__((ext_vector_type(8))) int i32x8;

#define NS 8           // shift channels
#define H 128
#define W 128
#define RT 16          // output rows per workgroup
#define TILE_H (RT + 8)    // 24
#define TILE_W (W + 8)     // 136 = 128 DWORDs + 8 DWORD pad (TDM pad feature)
#define NTAPS 90           // 10 ey values * 9 dx values
#define NCHUNK 23          // ceil(90/4) K-chunks of 4

// compile-time tap -> LDS element offset; taps >= NTAPS carry zero weight,
// so clamp them to a harmless in-range address.
#define TCLAMP(t) ((t) < NTAPS ? (t) : (NTAPS - 1))
#define OTF(t) ((TCLAMP(t) / 9) * TILE_W + (TCLAMP(t) % 9))

__device__ __forceinline__ float aw_eval(const float* __restrict__ filt,
                                         int sy, int sx, int s, int r, int t) {
    if (t >= NTAPS) return 0.0f;
    int ey = t / 9 - 4;       // row offset relative to base output row
    int dx = t % 9 - 4;       // col offset
    int p  = ey - r + sy;     // original filter row
    int q  = dx + sx;         // original filter col
    return (p >= 0 && p < 5 && q >= 0 && q < 5) ? filt[s * 25 + p * 5 + q] : 0.0f;
}

__global__ __launch_bounds__(256)
void shifted_conv_wmma(const float* __restrict__ tens,
                       const float* __restrict__ filt,
                       const int*   __restrict__ shifts,
                       float*       __restrict__ out)
{
    alignas(16) __shared__ float tile[TILE_H * TILE_W]; // input tile w/ 4-wide halo
    __shared__ v2f awgt[NCHUNK * 32];                   // per-lane A data per chunk

    const int n   = blockIdx.x;        // image index (B*C = 256)
    const int r0  = blockIdx.y * RT;   // output row-tile origin
    const int tid = threadIdx.x;
    const int wid = tid >> 5;          // 8 waves

    // ---- pre-zero LDS tile (provides top/left/right halo zeros) ----
    {
        u32x4 z = {0u, 0u, 0u, 0u};
        u32x4* tz = (u32x4*)tile;
        for (int idx = tid; idx < (TILE_H * TILE_W) / 4; idx += 256)
            tz[idx] = z;
    }

    // ---- build A (weights) for all 23 K-chunks ----
    // lane L: m = L&15 = s*2+r ; h = L>>4 ; VGPR0 = tap 4c+2h, VGPR1 = tap 4c+2h+1
    for (int idx = tid; idx < NCHUNK * 32; idx += 256) {
        int c    = idx >> 5;
        int lane = idx & 31;
        int m    = lane & 15;
        int hh   = lane >> 4;
        int s    = m >> 1;
        int r    = m & 1;
        int sy   = shifts[2 * s + 0];
        int sx   = shifts[2 * s + 1];
        int t0   = 4 * c + 2 * hh;
        v2f a;
        a.x = aw_eval(filt, sy, sx, s, r, t0);
        a.y = aw_eval(filt, sy, sx, s, r, t0 + 1);
        awgt[idx] = a;
    }

    __syncthreads();

    // ---- TDM: DMA the valid sub-rectangle of the image into the LDS tile ----
    // Rows [gy0, min(128, r0+20)) of width 128 land at LDS (row ly0, col 4).
    // LDS row stride 136 floats = tile rows of 128 DWORDs + TDM pad of 8 DWORDs
    // (pad_interval enum 6 = 128 DW, pad_amount enum 7 = 8 DW).
    // Bottom halo beyond the image comes from TDM OOB-reads-return-zero
    // (tile_dim1 rows requested vs tensor_dim1 rows available).
    if (wid == 0) {
        int gy0 = r0 - 4; if (gy0 < 0) gy0 = 0;
        int ly0 = gy0 - (r0 - 4);                 // 4 when r0==0, else 0
        uint32_t ldsb = (uint32_t)(uintptr_t)tile + (uint32_t)(ly0 * TILE_W + 4) * 4u;
        uint64_t ga   = (uint64_t)(uintptr_t)(tens + (size_t)n * (H * W) + (size_t)gy0 * W);
        uint32_t tdim1  = (uint32_t)(H - gy0);        // tensor rows from gy0
        uint32_t tiled1 = (uint32_t)(TILE_H - ly0);   // LDS rows to fill

        u32x4 g0;
        g0[0] = 1u;                                   // count=1 valid descriptor
        g0[1] = ldsb;                                 // lds_addr [63:32]
        g0[2] = (uint32_t)ga;                         // global_addr [95:64]
        g0[3] = (uint32_t)((ga >> 32) & 0x01FFFFFFu)  // global_addr [120:96]
              | (2u << 30);                           // type=2 ("image") [127:126]

        i32x8 g1;
        g1[0] = (int)((2u << 16)      // data_size = 4 bytes
              |       (1u << 20)      // pad_enable
              |       (6u << 22)      // pad_interval: 128 DWORDs
              |       (7u << 25));    // pad_amount: 8 DWORDs
        g1[1] = (int)((uint32_t)W << 16);     // tensor_dim0 low16 @ bits[63:48]
        g1[2] = (int)(tdim1 << 16);           // dim0 hi16=0 | tensor_dim1 low16
        g1[3] = (int)((uint32_t)W << 16);     // dim1 hi16=0 | tile_dim0 = 128
        g1[4] = (int)tiled1;                  // tile_dim1 | tile_dim2 = 0
        g1[5] = (int)W;                       // tensor_dim0_stride = 128 elems
        g1[6] = 0;
        g1[7] = 0;                            // tensor_dim1_stride unused (2D)

        i32x4 gz = {0, 0, 0, 0};
#if __clang_major__ >= 23
        i32x8 gz8 = {0, 0, 0, 0, 0, 0, 0, 0};
        __builtin_amdgcn_tensor_load_to_lds(g0, g1, gz, gz, gz8, 0);
#else
        __builtin_amdgcn_tensor_load_to_lds(g0, g1, gz, gz, 0);
#endif
        __builtin_amdgcn_s_wait_tensorcnt(0);
    }

    __syncthreads();

    const int lane = tid & 31;
    const int bcol = lane & 15;    // N column within 16-wide strip
    const int h    = lane >> 4;    // lane half
    const int j0   = wid * 16;     // column strip origin

    // hoist A chunks into registers (invariant across row-pairs)
    v2f aw[NCHUNK];
#pragma unroll
    for (int c = 0; c < NCHUNK; ++c) aw[c] = awgt[c * 32 + lane];

    const float* tbase = tile + j0 + bcol;

    for (int pair = 0; pair < RT / 2; ++pair) {
        const float* tp = tbase + pair * 2 * TILE_W;
        v8f acc0 = {};
        v8f acc1 = {};
#pragma unroll
        for (int c = 0; c < NCHUNK; ++c) {
            // per-half tap offsets are compile-time constants -> one cndmask each
            int ot0 = h ? OTF(4 * c + 2) : OTF(4 * c + 0);
            int ot1 = h ? OTF(4 * c + 3) : OTF(4 * c + 1);
            v2f bv;
            bv.x = tp[ot0];
            bv.y = tp[ot1];
            if (c & 1)
                acc1 = __builtin_amdgcn_wmma_f32_16x16x4_f32(
                    false, aw[c], false, bv, (short)0, acc1, false, false);
            else
                acc0 = __builtin_amdgcn_wmma_f32_16x16x4_f32(
                    false, aw[c], false, bv, (short)0, acc0, false, false);
        }
        v8f d = acc0 + acc1;

        // D layout: VGPR v, half h -> row m = v + 8h, col = bcol
        // out flat index: ((n*8 + s) * 128 + i) * 128 + j
#pragma unroll
        for (int v = 0; v < 8; ++v) {
            int m = v + 8 * h;
            int s = m >> 1;
            int r = m & 1;
            int gi = r0 + 2 * pair + r;
            size_t o = ((size_t)(n * NS + s) << 14) + (size_t)gi * 128 + (j0 + bcol);
            out[o] = d[v];
        }
    }
}

extern "C" void kernel_launch(void* const* d_in, const int* in_sizes, int n_in,
                              void* d_out, int out_size, void* d_ws, size_t ws_size,
                              hipStream_t stream) {
    const float* tens   = (const float*)d_in[0];   // (16,16,128,128) fp32
    const float* filt   = (const float*)d_in[1];   // (8,1,5,5) fp32
    const int*   shifts = (const int*)d_in[2];     // (8,2) int32
    float*       out    = (float*)d_out;           // (16,128,128,128) fp32

    dim3 grid(256, H / RT);   // 256 images x 8 row tiles
    dim3 block(256);          // 8 wave32
    shifted_conv_wmma<<<grid, block, 0, stream>>>(tens, filt, shifts, out);
}